// Contrast_gene_26371099197446
// MI455X (gfx1250) — compile-verified
//
#include <hip/hip_runtime.h>
#include <math.h>

namespace {
constexpr int   D              = 128;
constexpr float TAU            = 0.5f;
constexpr float EPS            = 1e-8f;
constexpr int   BLOCK          = 256;                    // 8 wave32s
constexpr int   TILE_ROWS      = 16;
constexpr int   TILE_FLOATS    = TILE_ROWS * D;          // 2048 floats = 8 KB per array
constexpr int   CHUNKS         = TILE_FLOATS / (BLOCK * 4); // 2 float4 chunks / thread / array
constexpr int   LOADS_PER_TILE = 3 * CHUNKS;             // 6 async b128 per thread per tile
constexpr int   NBLOCKS        = 640;
static_assert(LOADS_PER_TILE == 6, "wait immediates below assume 6");
}

typedef int   v4i __attribute__((ext_vector_type(4)));
typedef float v2f __attribute__((ext_vector_type(2)));
typedef float v8f __attribute__((ext_vector_type(8)));
typedef __attribute__((address_space(1))) v4i* v4i_gptr;
typedef __attribute__((address_space(3))) v4i* v4i_lptr;

// ---- CDNA5 async global->LDS copy (ASYNCcnt-tracked) ------------------------
__device__ __forceinline__ void async_load_f4(const float* g, float* l) {
  v4i* gg = (v4i*)(void*)g;   // drop const / retype (generic)
  v4i* ll = (v4i*)l;
  __builtin_amdgcn_global_load_async_to_lds_b128(
      (v4i_gptr)gg, (v4i_lptr)ll, /*offset=*/0, /*cpol=*/0);
}

template <int N>
__device__ __forceinline__ void wait_async() {
#if __has_builtin(__builtin_amdgcn_s_wait_asynccnt)
  __builtin_amdgcn_s_wait_asynccnt(N);
#else
  if constexpr (N == 0) asm volatile("s_wait_asynccnt 0" ::: "memory");
  else                  asm volatile("s_wait_asynccnt 6" ::: "memory");
#endif
}

// ---- Stage 1: per-block partial sums of ss = pos/(neg+eps) ------------------
__global__ __launch_bounds__(BLOCK) void contrast_partials_kernel(
    const float* __restrict__ xr, const float* __restrict__ xp,
    const float* __restrict__ xn, float* __restrict__ partials, int n)
{
  __shared__ __align__(16) float tile_lds[2][3][TILE_FLOATS];   // 48 KB, double buffered
  __shared__ float wave_sums[BLOCK / 32];

  const int tid  = (int)threadIdx.x;
  const int lane = tid & 31;
  const int wave = tid >> 5;
  const long long total_f  = (long long)n * D;
  const int       num_tiles = (n + TILE_ROWS - 1) / TILE_ROWS;
  const float* const srcs[3] = {xr, xp, xn};

  auto issue_tile = [&](int tileIdx, int buf) {
    const long long base = (long long)tileIdx * TILE_FLOATS;
#pragma unroll
    for (int a = 0; a < 3; ++a) {
#pragma unroll
      for (int k = 0; k < CHUNKS; ++k) {
        const int c = tid + k * BLOCK;               // contiguous chunk id -> coalesced
        long long gf = base + (long long)c * 4;
        if (gf + 4 > total_f) gf = total_f - 4;      // clamped dummy; such rows unused
        async_load_f4(srcs[a] + gf, &tile_lds[buf][a][c * 4]);
      }
    }
  };

  float acc = 0.0f;
  int buf  = 0;
  int tile = (int)blockIdx.x;
  if (tile < num_tiles) issue_tile(tile, buf);       // prologue prefetch

  for (; tile < num_tiles; tile += (int)gridDim.x) {
    const int next = tile + (int)gridDim.x;
    if (next < num_tiles) {
      issue_tile(next, buf ^ 1);                     // prefetch next tile
      wait_async<LOADS_PER_TILE>();                  // in-order: current tile done
    } else {
      wait_async<0>();
    }
    __syncthreads();                                 // all waves' async data visible

    // 8 waves x 2 rows = 16 rows; one wave owns a full 128-float row (32 x float4)
#pragma unroll
    for (int rr = 0; rr < TILE_ROWS / (BLOCK / 32); ++rr) {
      const int r = wave + rr * (BLOCK / 32);
      const long long row = (long long)tile * TILE_ROWS + r;
      if (row < (long long)n) {
        const int o = r * D + lane * 4;
        const float4 vr = *(const float4*)&tile_lds[buf][0][o];
        const float4 vp = *(const float4*)&tile_lds[buf][1][o];
        const float4 vn = *(const float4*)&tile_lds[buf][2][o];
        float s_rp = vr.x*vp.x + vr.y*vp.y + vr.z*vp.z + vr.w*vp.w;
        float s_rn = vr.x*vn.x + vr.y*vn.y + vr.z*vn.z + vr.w*vn.w;
        float s_rr = vr.x*vr.x + vr.y*vr.y + vr.z*vr.z + vr.w*vr.w;
        float s_pp = vp.x*vp.x + vp.y*vp.y + vp.z*vp.z + vp.w*vp.w;
        float s_nn = vn.x*vn.x + vn.y*vn.y + vn.z*vn.z + vn.w*vn.w;
#pragma unroll
        for (int m = 16; m >= 1; m >>= 1) {          // wave32 xor-tree reduction
          s_rp += __shfl_xor(s_rp, m, 32);
          s_rn += __shfl_xor(s_rn, m, 32);
          s_rr += __shfl_xor(s_rr, m, 32);
          s_pp += __shfl_xor(s_pp, m, 32);
          s_nn += __shfl_xor(s_nn, m, 32);
        }
        const float nr  = sqrtf(s_rr);
        const float pos = expf(s_rp / (nr * sqrtf(s_pp)) / TAU);
        const float neg = expf(s_rn / (nr * sqrtf(s_nn)) / TAU);
        acc += pos / (neg + EPS);                    // identical in all lanes
      }
    }
    __syncthreads();                                 // done reading buf before reuse
    buf ^= 1;
  }

  if (lane == 0) wave_sums[wave] = acc;
  __syncthreads();
  if (tid == 0) {
    float s = 0.0f;
#pragma unroll
    for (int w = 0; w < BLOCK / 32; ++w) s += wave_sums[w];  // fixed order
    partials[blockIdx.x] = s;
  }
}

// ---- Stage 2: WMMA-based final reduction, loss = -log(sum) ------------------
// V_WMMA_F32_16X16X4_F32 with B = all-ones: D[m][n] = sum_k A[m][k].
// Addition is permutation-invariant, so any bijective packing of 64 partials
// into A's (M,K) slots is exact:  a.x = p[base+lane], a.y = p[base+32+lane].
// Single wave32; EXEC all-ones; no divergence around the WMMA.
__global__ __launch_bounds__(32) void contrast_final_kernel(
    const float* __restrict__ partials, int nb, float* __restrict__ out)
{
  const int lane = (int)threadIdx.x;                 // 0..31
  v8f c = {0.f, 0.f, 0.f, 0.f, 0.f, 0.f, 0.f, 0.f};
  v2f ones; ones.x = 1.0f; ones.y = 1.0f;            // every B slot = 1, layout-oblivious

  for (int base = 0; base < nb; base += 64) {        // uniform trip count: no divergence
    const int i0 = base + lane;
    const int i1 = base + 32 + lane;
    v2f a;
    a.x = (i0 < nb) ? partials[i0] : 0.0f;
    a.y = (i1 < nb) ? partials[i1] : 0.0f;
    c = __builtin_amdgcn_wmma_f32_16x16x4_f32(
        /*neg_a=*/false, a, /*neg_b=*/false, ones,
        /*c_mod=*/(short)0, c, /*reuse_a=*/false, /*reuse_b=*/false);
  }

  // Lane l holds D[m][l%16] for 8 rows m; combine the two row-halves.
  float t = c[0] + c[1] + c[2] + c[3] + c[4] + c[5] + c[6] + c[7];
  t += __shfl_xor(t, 16, 32);
  if (lane == 0) out[0] = -logf(t);
}

extern "C" void kernel_launch(void* const* d_in, const int* in_sizes, int n_in,
                              void* d_out, int out_size, void* d_ws, size_t ws_size,
                              hipStream_t stream) {
  const float* xr = (const float*)d_in[0];
  const float* xp = (const float*)d_in[1];
  const float* xn = (const float*)d_in[2];
  float* out = (float*)d_out;
  const int n = in_sizes[0] / D;

  int nb = NBLOCKS;
  if (ws_size < (size_t)nb * sizeof(float)) nb = (int)(ws_size / sizeof(float));
  if (nb < 1) nb = 1;
  float* partials = (float*)d_ws;

  contrast_partials_kernel<<<dim3(nb), dim3(BLOCK), 0, stream>>>(xr, xp, xn, partials, n);
  contrast_final_kernel<<<dim3(1), dim3(32), 0, stream>>>(partials, nb, out);
}